// MOCA_38637525794939
// MI455X (gfx1250) — compile-verified
//
#include <hip/hip_runtime.h>
#include <cstdint>

// Problem shape (fixed by the reference setup_inputs()).
#define BB 32
#define TT 512
#define DD 16
#define RR (TT + 1)            // 513 run-lengths
#define BLOCK 544              // 17 waves of 32; thread tid owns run-length r=tid (tid<=512)
#define NWAVES (BLOCK / 32)    // 17
#define CHUNK 16               // timesteps staged per TDM transfer
#define NCHUNK (TT / CHUNK)    // 32

#define SIGMA2f 0.25f
#define INV_SIGMA2f 4.0f
#define LOG2PIf 1.837877066409345f

#if __has_builtin(__builtin_amdgcn_tensor_load_to_lds) && __has_builtin(__builtin_amdgcn_s_wait_tensorcnt)
#define HAVE_TDM 1
#else
#define HAVE_TDM 0
#endif

typedef __attribute__((ext_vector_type(4))) unsigned int u32x4;
typedef __attribute__((ext_vector_type(8))) int i32x8;
typedef __attribute__((ext_vector_type(4))) int i32x4;

__device__ __forceinline__ unsigned lds_byte_off(const void* p) {
  // Generic pointer to LDS: addr[31:0] is the LDS byte offset (aperture form).
  return (unsigned)(uintptr_t)p;
}

// Hardware transcendentals (~1 ulp): v_rcp_f32 / v_log_f32 / native exp.
__device__ __forceinline__ float fast_rcp(float x) { return __builtin_amdgcn_rcpf(x); }

#if HAVE_TDM
// 1-D contiguous tile load via the Tensor Data Mover (TENSORcnt-tracked DMA).
// D# per CDNA5 ISA ch.8: group0 = {flags, lds_addr, global_addr, type=2},
// group1 = {data_size=4B, tensor_dim0=nelem, tensor_dim1=1, tile_dim0=nelem}.
__device__ __forceinline__ void tdm_load_row(unsigned lds_byte, const float* gsrc, int nelem) {
  unsigned long long ga = (unsigned long long)(uintptr_t)gsrc;
  u32x4 g0;
  g0[0] = 1u;                                            // count=1, user descriptor
  g0[1] = lds_byte;                                      // lds_addr [63:32]
  g0[2] = (unsigned)(ga & 0xFFFFFFFFull);                // global_addr [95:64]
  g0[3] = (unsigned)((ga >> 32) & 0x1FFFFFFull)          // global_addr [120:96]
        | (2u << 30);                                    // type=2 ("image")
  i32x8 g1 = {0, 0, 0, 0, 0, 0, 0, 0};
  g1[0] = 0x00020000;                                    // data_size=2 -> 4 bytes
  g1[1] = (int)((nelem & 0xFFFF) << 16);                 // tensor_dim0 lo16 @bits63:48
  g1[2] = (int)(((unsigned)nelem >> 16) & 0xFFFFu)       // tensor_dim0 hi16
        | (1 << 16);                                     // tensor_dim1 = 1
  g1[3] = (int)((nelem & 0xFFFF) << 16);                 // tile_dim0 @bits127:112
  g1[5] = nelem;                                         // tensor_dim0_stride lo32
  i32x4 z4 = {0, 0, 0, 0};
#if defined(__clang_major__) && (__clang_major__ >= 23)
  i32x8 z8 = {0, 0, 0, 0, 0, 0, 0, 0};
  __builtin_amdgcn_tensor_load_to_lds(g0, g1, z4, z4, z8, 0);
#else
  __builtin_amdgcn_tensor_load_to_lds(g0, g1, z4, z4, 0);
#endif
}
#endif

// Online logsumexp pair combine: (m,s) represents log(s) + m with s >= 0.
__device__ __forceinline__ void lse_combine(float& m, float& s, float m2, float s2) {
  if (m2 == -__builtin_inff()) return;
  if (m == -__builtin_inff()) { m = m2; s = s2; return; }
  if (m2 > m) { float tm = m; m = m2; m2 = tm; float ts = s; s = s2; s2 = ts; }
  s += s2 * __expf(m2 - m);
}

__global__ __launch_bounds__(BLOCK, 1) void moca_scan_kernel(
    const float* __restrict__ x_mat, const float* __restrict__ y_mat,
    const float* __restrict__ hazard_logit,
    const float* __restrict__ q0, const float* __restrict__ lam0,
    float* __restrict__ out)
{
  __shared__ float xb[2][CHUNK * DD];
  __shared__ float yb[2][CHUNK * DD];
  __shared__ float red_m[NWAVES];
  __shared__ float red_s[NWAVES];
  __shared__ float bcast_lse;

  const int b    = blockIdx.x;
  const int tid  = (int)threadIdx.x;
  const int lane = tid & 31;
  const int wid  = tid >> 5;
  const int r    = tid;              // run-length owned by this thread
  const bool rv  = (tid < RR);       // tid <= 512 valid; 513..543 are barrier fillers
  const float NEG = -__builtin_inff();

  // Output sections (flat, reference return order).
  float* nll_out  = out;                                   // [B,T]
  float* hist_out = out + (size_t)BB * TT;                 // [B,T,R]
  float* q_out    = hist_out + (size_t)BB * TT * RR;       // [B,R,D]
  float* lam_out  = q_out + (size_t)BB * RR * DD;          // [B,R,D]
  float* lpf_out  = lam_out + (size_t)BB * RR * DD;        // [B,R]

  // log_sigmoid(z) and log_sigmoid(-z): computed once, keep precise math.
  const float z = hazard_logit[0];
  const float log_h   = (z > 0.f)  ? -log1pf(expf(-z)) : (z  - log1pf(expf(z)));
  const float zn = -z;
  const float log_1mh = (zn > 0.f) ? -log1pf(expf(-zn)) : (zn - log1pf(expf(zn)));

  // Register-resident per-run-length state.
  float q[DD], lam[DD];
#pragma unroll
  for (int d = 0; d < DD; ++d) { q[d] = q0[d]; lam[d] = lam0[d]; }
  float lp = (r == 0) ? 0.f : NEG;   // log P(r | x_{1:t})

  const float* xg = x_mat + (size_t)b * TT * DD;
  const float* yg = y_mat + (size_t)b * TT * DD;

#if HAVE_TDM
  if (wid == 0) {  // wave 0 drives the DMA pipeline
    tdm_load_row(lds_byte_off(&xb[0][0]), xg, CHUNK * DD);
    tdm_load_row(lds_byte_off(&yb[0][0]), yg, CHUNK * DD);
  }
#endif

  for (int c = 0; c < NCHUNK; ++c) {
    const int cb = c & 1;
#if HAVE_TDM
    if (wid == 0) {
      if (c + 1 < NCHUNK) {
        const int nb = (c + 1) & 1;
        tdm_load_row(lds_byte_off(&xb[nb][0]), xg + (size_t)(c + 1) * CHUNK * DD, CHUNK * DD);
        tdm_load_row(lds_byte_off(&yb[nb][0]), yg + (size_t)(c + 1) * CHUNK * DD, CHUNK * DD);
        __builtin_amdgcn_s_wait_tensorcnt(2);   // current chunk's 2 DMAs done; next 2 in flight
      } else {
        __builtin_amdgcn_s_wait_tensorcnt(0);
      }
    }
#else
    for (int i = tid; i < CHUNK * DD; i += BLOCK) {
      xb[cb][i] = xg[(size_t)c * CHUNK * DD + i];
      yb[cb][i] = yg[(size_t)c * CHUNK * DD + i];
    }
#endif
    __syncthreads();   // publish staged chunk to all waves

    for (int i = 0; i < CHUNK; ++i) {
      const int t = c * CHUNK + i;
      const float* xv = &xb[cb][i * DD];
      const float* yv = &yb[cb][i * DD];

      // log_prgx_hist stores the prior (carry at entry of the step).
      if (rv) hist_out[((size_t)b * TT + t) * RR + r] = lp;

      float joint;
      if (rv && r <= t) {
        float acc = 0.f;
#pragma unroll
        for (int d = 0; d < DD; ++d) {
          const float xd = xv[d], yd = yv[d];
          const float x2 = xd * xd;
          const float rl = fast_rcp(lam[d]);       // old lambda (v_rcp_f32)
          const float mean = xd * q[d] * rl;       // old q
          const float var  = SIGMA2f + x2 * rl;
          const float diff = yd - mean;
          acc += diff * diff * fast_rcp(var) + __logf(var);
          q[d]   += xd * yd * INV_SIGMA2f;         // update after use (valid: r<=t)
          lam[d] += x2 * INV_SIGMA2f;
        }
        joint = -0.5f * (acc + DD * LOG2PIf) + lp;
      } else {
        joint = NEG;                               // lp == -inf for r > t anyway
      }

      // Block-wide logsumexp over joint: wave32 butterfly, then 17 partials.
      float m = joint, s = 1.0f;
#pragma unroll
      for (int off = 16; off >= 1; off >>= 1) {
        const float m2 = __shfl_xor(m, off, 32);
        const float s2 = __shfl_xor(s, off, 32);
        lse_combine(m, s, m2, s2);
      }
      if (lane == 0) { red_m[wid] = m; red_s[wid] = s; }
      __syncthreads();
      if (wid == 0) {
        float mm = (lane < NWAVES) ? red_m[lane] : NEG;
        float ss = (lane < NWAVES) ? red_s[lane] : 0.0f;
#pragma unroll
        for (int off = 16; off >= 1; off >>= 1) {
          const float m2 = __shfl_xor(mm, off, 32);
          const float s2 = __shfl_xor(ss, off, 32);
          lse_combine(mm, ss, m2, s2);
        }
        if (lane == 0) bcast_lse = mm + __logf(ss);
      }
      __syncthreads();
      const float lse = bcast_lse;

      if (tid == 0) nll_out[(size_t)b * TT + t] = -lse;

      // log_softmax(log_prx) == log_prx - lse  (exact: lse(log_prx) = lse(joint)).
      if (rv) lp = (r == t + 1) ? log_h : (log_1mh + joint - lse);
      // (r <= t: finite; r > t+1: joint = -inf -> stays -inf.)
    }
  }

  // Final state outputs.
  if (rv) {
#pragma unroll
    for (int d = 0; d < DD; ++d) {
      q_out[((size_t)b * RR + r) * DD + d]   = q[d];
      lam_out[((size_t)b * RR + r) * DD + d] = lam[d];
    }
    lpf_out[(size_t)b * RR + r] = lp;
  }
}

extern "C" void kernel_launch(void* const* d_in, const int* in_sizes, int n_in,
                              void* d_out, int out_size, void* d_ws, size_t ws_size,
                              hipStream_t stream) {
  (void)in_sizes; (void)n_in; (void)out_size; (void)d_ws; (void)ws_size;
  const float* x_mat = (const float*)d_in[0];
  const float* y_mat = (const float*)d_in[1];
  const float* hz    = (const float*)d_in[2];
  const float* q0    = (const float*)d_in[3];
  const float* lam0  = (const float*)d_in[4];
  float* out = (float*)d_out;
  moca_scan_kernel<<<dim3(BB), dim3(BLOCK), 0, stream>>>(x_mat, y_mat, hz, q0, lam0, out);
}